// Edge_Cycle_39745627357898
// MI455X (gfx1250) — compile-verified
//
#include <hip/hip_runtime.h>

// ---------------------------------------------------------------------------
// Edge/Cycle ptens block for MI455X (gfx1250, wave32, WMMA).
//
//  1. build cycle_new [73000 x 640] from edge_rep/cycle_rep (closed-form ring
//     structure, no atomics -> deterministic).
//  2. Y1c = cycle_new @ Wc1            (WMMA f32 16x16x4 GEMM, mode 0)
//  3. Y1e = gather(edge_in) @ We1      (gather fused into A-tile load, mode 1)
//  4. column mean/var (two-phase deterministic reduction) -> scale/shift
//  5. Y2 = bnrelu(Y1) @ W2             (BN+relu fused into A-tile load, mode 2)
//     written straight into d_out, then BN+relu applied in place.
//
// Staging uses B128 global loads; B128 LDS stores for A, scalar for B (16B-aligned padded
// LDS rows, stride 36 floats).
// ---------------------------------------------------------------------------

typedef __attribute__((ext_vector_type(2))) float v2f;
typedef __attribute__((ext_vector_type(4))) float v4f;
typedef __attribute__((ext_vector_type(8))) float v8f;

#define N_HEX   8000
#define N_PENT  5000
#define HCH     128
#define N_C5    (5 * N_PENT)        /* 25000 */
#define N_C6    (6 * N_HEX)         /* 48000 */
#define N_CYC   (N_C5 + N_C6)       /* 73000 */
#define N_EDGES (6 * N_HEX + 5 * N_PENT) /* 73000 */
#define N_EROWS (2 * N_EDGES)       /* 146000 */
#define CN_W    640                 /* 5*H */
#define EIN_W   1408                /* 11*H */
#define HID     256                 /* 2*H */
#define STATS_G 1024
#define LDA_PAD 36                  /* 144B rows: 16B aligned, conflict-free */

// ---------------------------------------------------------------------------
// Stage 1: build cycle_new.
// Row layout: [copy(128)|bcast(128)|ringsum_copy(128)|ringsum_bcast(128)|cycle_rep(128)]
// ---------------------------------------------------------------------------
__global__ void build_cn_pent(const float* __restrict__ edge_rep,
                              const float* __restrict__ cycle_rep,
                              float* __restrict__ cn) {
  int p = blockIdx.x;          // pentagon id
  int c = threadIdx.x;         // channel 0..127
  float A0[5], A1[5];
  for (int q = 0; q < 5; ++q) {
    int e = 6 * N_HEX + p * 5 + q;            // global edge id
    A0[q] = edge_rep[(size_t)(2 * e) * HCH + c];
    A1[q] = edge_rep[(size_t)(2 * e + 1) * HCH + c];
  }
  float cp[5], bc[5], rsC = 0.f, rsB = 0.f;
  for (int q = 0; q < 5; ++q) {
    int qm = (q == 0) ? 4 : q - 1;
    cp[q] = A0[q] + A1[qm];
    bc[q] = (A0[q] + A1[q]) + (A0[qm] + A1[qm]);
    rsC += cp[q]; rsB += bc[q];
  }
  for (int q = 0; q < 5; ++q) {
    size_t t = (size_t)p * 5 + q;             // cycle row (c5 block first)
    float* o = cn + t * CN_W;
    o[c]           = cp[q];
    o[HCH + c]     = bc[q];
    o[2 * HCH + c] = rsC;
    o[3 * HCH + c] = rsB;
    o[4 * HCH + c] = cycle_rep[t * HCH + c];
  }
}

__global__ void build_cn_hex(const float* __restrict__ edge_rep,
                             const float* __restrict__ cycle_rep,
                             float* __restrict__ cn) {
  int h = blockIdx.x;
  int c = threadIdx.x;
  float A0[6], A1[6];
  for (int k = 0; k < 6; ++k) {
    int e = h * 6 + k;
    A0[k] = edge_rep[(size_t)(2 * e) * HCH + c];
    A1[k] = edge_rep[(size_t)(2 * e + 1) * HCH + c];
  }
  float cp[6], bc[6], rsC = 0.f, rsB = 0.f;
  for (int k = 0; k < 6; ++k) {
    int km = (k == 0) ? 5 : k - 1;
    cp[k] = A0[k] + A1[km];
    bc[k] = (A0[k] + A1[k]) + (A0[km] + A1[km]);
    rsC += cp[k]; rsB += bc[k];
  }
  for (int k = 0; k < 6; ++k) {
    size_t t = (size_t)N_C5 + (size_t)h * 6 + k;
    float* o = cn + t * CN_W;
    o[c]           = cp[k];
    o[HCH + c]     = bc[k];
    o[2 * HCH + c] = rsC;
    o[3 * HCH + c] = rsB;
    o[4 * HCH + c] = cycle_rep[t * HCH + c];
  }
}

// ---------------------------------------------------------------------------
// WMMA GEMM: Y[M x N] = A'[M x K] @ B[K x N]
//   MODE 0: A' = A (dense, leading dim lda)
//   MODE 1: A' = gathered edge_in row:
//             cols [0,128)    = edge_rep[r]
//             cols [128,768)  = cn[src_self(r)]
//             cols [768,1408) = cn[src_a(e)] + cn[src_b(e)]
//   MODE 2: A' = relu(A * scale[col] + shift[col])   (fused batch-norm)
// Block: 256 threads = 8 waves; tile 128(M) x 64(N) x 32(K); each wave owns a
// 32x32 sub-tile as 2x2 v_wmma_f32_16x16x4_f32 accumulators.
// ---------------------------------------------------------------------------
template <int MODE>
__global__ void gemm_wmma(const float* __restrict__ A,
                          const float* __restrict__ B,
                          float* __restrict__ Y,
                          int M, int N, int K, int lda,
                          const float* __restrict__ scale,
                          const float* __restrict__ shift,
                          const float* __restrict__ edge_rep,
                          const float* __restrict__ cn) {
  __shared__ __align__(16) float As[128][LDA_PAD];
  __shared__ __align__(16) float Bs[64][LDA_PAD];   // transposed: Bs[n][k]
  __shared__ int rowSelf[128], rowA[128], rowB[128];

  const int tid  = threadIdx.x;
  const int row0 = blockIdx.x * 128;
  const int col0 = blockIdx.y * 64;

  if (MODE == 1) {
    if (tid < 128) {
      int rg = row0 + tid;
      int sS = 0, sA = 0, sB = 0;
      if (rg < M) {
        int e = rg >> 1;
        if (e < N_C6) {                        // hexagon edge
          int h = e / 6, k = e - h * 6;
          int kn = (k + 1 == 6) ? 0 : k + 1;
          sA = N_C5 + e;
          sB = N_C5 + h * 6 + kn;
        } else {                               // pentagon edge
          int ep = e - N_C6;
          int p = ep / 5, q = ep - p * 5;
          int qn = (q + 1 == 5) ? 0 : q + 1;
          sA = ep;
          sB = p * 5 + qn;
        }
        sS = (rg & 1) ? sB : sA;
      }
      rowSelf[tid] = sS; rowA[tid] = sA; rowB[tid] = sB;
    }
    __syncthreads();
  }

  float acc[2][2][8];
#pragma unroll
  for (int i = 0; i < 2; ++i)
#pragma unroll
    for (int j = 0; j < 2; ++j)
#pragma unroll
      for (int v = 0; v < 8; ++v) acc[i][j][v] = 0.f;

  const int wave = tid >> 5, lane = tid & 31;
  const int wm = wave & 3, wn = wave >> 2;       // 4x2 wave grid
  const int lrow = lane & 15;
  const int lk   = (lane >> 4) << 1;             // 0 or 2

  for (int k0 = 0; k0 < K; k0 += 32) {
    // --- stage A tile (128 x 32) with B128 loads/stores ---
#pragma unroll
    for (int i = 0; i < 4; ++i) {
      int flat4 = tid + 256 * i;                 // 1024 float4s
      int r  = flat4 >> 3;                       // 8 float4 per row
      int c4 = (flat4 & 7) << 2;                 // col 0,4,...,28
      int rg = row0 + r, gc = k0 + c4;
      v4f v = {0.f, 0.f, 0.f, 0.f};
      if (rg < M) {
        if (MODE == 0) {
          v = *(const v4f*)&A[(size_t)rg * lda + gc];
        } else if (MODE == 2) {
          v4f y  = *(const v4f*)&A[(size_t)rg * lda + gc];
          v4f sc = *(const v4f*)&scale[gc];
          v4f sh = *(const v4f*)&shift[gc];
          v4f t  = y * sc + sh;
#pragma unroll
          for (int u = 0; u < 4; ++u) v[u] = t[u] > 0.f ? t[u] : 0.f;
        } else { // MODE == 1 : fused gather (region boundaries 128/768 are
                 // multiples of 32, so each chunk hits exactly one region)
          if (gc < 128) {
            v = *(const v4f*)&edge_rep[(size_t)rg * HCH + gc];
          } else if (gc < 768) {
            v = *(const v4f*)&cn[(size_t)rowSelf[r] * CN_W + (gc - 128)];
          } else {
            v4f a = *(const v4f*)&cn[(size_t)rowA[r] * CN_W + (gc - 768)];
            v4f b = *(const v4f*)&cn[(size_t)rowB[r] * CN_W + (gc - 768)];
            v = a + b;
          }
        }
      }
      *(v4f*)&As[r][c4] = v;
    }
    // --- stage B tile (32 x 64): B128 global load, transposed LDS store ---
#pragma unroll
    for (int i = 0; i < 2; ++i) {
      int flat4 = tid + 256 * i;                 // 512 float4s
      int kr = flat4 >> 4;                       // 16 float4 per K-row
      int c4 = (flat4 & 15) << 2;
      v4f b = *(const v4f*)&B[(size_t)(k0 + kr) * N + (col0 + c4)];
#pragma unroll
      for (int u = 0; u < 4; ++u) Bs[c4 + u][kr] = b[u];
    }
    __syncthreads();

    // --- WMMA over the K-chunk ---
#pragma unroll
    for (int kk = 0; kk < 32; kk += 4) {
      v2f a0 = *(const v2f*)&As[wm * 32 + lrow][kk + lk];
      v2f a1 = *(const v2f*)&As[wm * 32 + 16 + lrow][kk + lk];
      v2f b0 = *(const v2f*)&Bs[wn * 32 + lrow][kk + lk];
      v2f b1 = *(const v2f*)&Bs[wn * 32 + 16 + lrow][kk + lk];
      v8f c00 = *(v8f*)acc[0][0];
      v8f c01 = *(v8f*)acc[0][1];
      v8f c10 = *(v8f*)acc[1][0];
      v8f c11 = *(v8f*)acc[1][1];
      c00 = __builtin_amdgcn_wmma_f32_16x16x4_f32(false, a0, false, b0,
                                                  (short)0, c00, false, false);
      c01 = __builtin_amdgcn_wmma_f32_16x16x4_f32(false, a0, false, b1,
                                                  (short)0, c01, false, false);
      c10 = __builtin_amdgcn_wmma_f32_16x16x4_f32(false, a1, false, b0,
                                                  (short)0, c10, false, false);
      c11 = __builtin_amdgcn_wmma_f32_16x16x4_f32(false, a1, false, b1,
                                                  (short)0, c11, false, false);
      *(v8f*)acc[0][0] = c00;
      *(v8f*)acc[0][1] = c01;
      *(v8f*)acc[1][0] = c10;
      *(v8f*)acc[1][1] = c11;
    }
    __syncthreads();
  }

  // --- store D tiles: VGPR v -> row v + 8*(lane/16); col = lane%16 ---
  const int drow = (lane >> 4) * 8;
#pragma unroll
  for (int i = 0; i < 2; ++i)
#pragma unroll
    for (int j = 0; j < 2; ++j)
#pragma unroll
      for (int v = 0; v < 8; ++v) {
        int rg = row0 + wm * 32 + i * 16 + drow + v;
        int cg = col0 + wn * 32 + j * 16 + lrow;
        if (rg < M) Y[(size_t)rg * N + cg] = acc[i][j][v];
      }
}

// ---------------------------------------------------------------------------
// Deterministic column statistics (two-phase, fixed reduction order).
// ---------------------------------------------------------------------------
__global__ void stats_partial(const float* __restrict__ Yb, int M, int ncols,
                              float* __restrict__ part) {
  int c = threadIdx.x;               // blockDim.x == ncols
  float s = 0.f, s2 = 0.f;
  for (int r = blockIdx.x; r < M; r += STATS_G) {
    float x = Yb[(size_t)r * ncols + c];
    s += x; s2 += x * x;
  }
  part[(size_t)blockIdx.x * 2 * ncols + c] = s;
  part[(size_t)blockIdx.x * 2 * ncols + ncols + c] = s2;
}

__global__ void stats_finalize(const float* __restrict__ part, int ncols, int M,
                               const float* __restrict__ gamma,
                               const float* __restrict__ beta,
                               float* __restrict__ scale,
                               float* __restrict__ shift) {
  int c = threadIdx.x;               // blockDim.x == ncols
  float s = 0.f, s2 = 0.f;
  for (int g = 0; g < STATS_G; ++g) {
    s  += part[(size_t)g * 2 * ncols + c];
    s2 += part[(size_t)g * 2 * ncols + ncols + c];
  }
  float mean = s / (float)M;
  float var  = s2 / (float)M - mean * mean;
  float inv  = rsqrtf(var + 1e-5f);
  float sc   = gamma[c] * inv;
  scale[c] = sc;
  shift[c] = beta[c] - mean * sc;
}

__global__ void bn_relu_apply(float* __restrict__ Yb, size_t total4, int ncols,
                              const float* __restrict__ scale,
                              const float* __restrict__ shift) {
  size_t i4 = (size_t)blockIdx.x * blockDim.x + threadIdx.x;
  if (i4 >= total4) return;
  int c = (int)((i4 << 2) % (size_t)ncols);      // ncols divisible by 4
  v4f y  = *(v4f*)&Yb[i4 << 2];
  v4f sc = *(const v4f*)&scale[c];
  v4f sh = *(const v4f*)&shift[c];
  v4f t  = y * sc + sh;
#pragma unroll
  for (int u = 0; u < 4; ++u) t[u] = t[u] > 0.f ? t[u] : 0.f;
  *(v4f*)&Yb[i4 << 2] = t;
}

// ---------------------------------------------------------------------------
// Host launch
// ---------------------------------------------------------------------------
extern "C" void kernel_launch(void* const* d_in, const int* in_sizes, int n_in,
                              void* d_out, int out_size, void* d_ws, size_t ws_size,
                              hipStream_t stream) {
  const float* edge_rep  = (const float*)d_in[0];
  const float* cycle_rep = (const float*)d_in[1];
  const float* We1 = (const float*)d_in[2];
  const float* ge1 = (const float*)d_in[3];
  const float* be1 = (const float*)d_in[4];
  const float* We2 = (const float*)d_in[5];
  const float* ge2 = (const float*)d_in[6];
  const float* be2 = (const float*)d_in[7];
  const float* Wc1 = (const float*)d_in[8];
  const float* gc1 = (const float*)d_in[9];
  const float* bc1 = (const float*)d_in[10];
  const float* Wc2 = (const float*)d_in[11];
  const float* gc2 = (const float*)d_in[12];
  const float* bc2 = (const float*)d_in[13];
  (void)in_sizes; (void)n_in; (void)out_size; (void)ws_size;

  // workspace layout (floats)
  float* ws  = (float*)d_ws;
  float* cn   = ws;                                     // 73000*640
  float* Y1e  = cn  + (size_t)N_CYC * CN_W;             // 146000*256
  float* Y1c  = Y1e + (size_t)N_EROWS * HID;            // 73000*256
  float* part = Y1c + (size_t)N_CYC * HID;              // 1024*2*256
  float* ss   = part + (size_t)STATS_G * 2 * 256;       // 8*256
  float* sc_e1 = ss,         * sh_e1 = ss + 256;
  float* sc_c1 = ss + 512,   * sh_c1 = ss + 768;
  float* sc_e2 = ss + 1024,  * sh_e2 = ss + 1280;
  float* sc_c2 = ss + 1536,  * sh_c2 = ss + 1792;

  float* outE = (float*)d_out;                          // 146000*128
  float* outC = outE + (size_t)N_EROWS * HCH;           // 73000*128

  // 1. cycle_new
  build_cn_pent<<<N_PENT, HCH, 0, stream>>>(edge_rep, cycle_rep, cn);
  build_cn_hex <<<N_HEX,  HCH, 0, stream>>>(edge_rep, cycle_rep, cn);

  // 2. layer-1 GEMMs
  dim3 blk(256);
  dim3 g1c((N_CYC + 127) / 128, HID / 64);
  gemm_wmma<0><<<g1c, blk, 0, stream>>>(cn, Wc1, Y1c, N_CYC, HID, CN_W, CN_W,
                                        nullptr, nullptr, nullptr, nullptr);
  dim3 g1e((N_EROWS + 127) / 128, HID / 64);
  gemm_wmma<1><<<g1e, blk, 0, stream>>>(nullptr, We1, Y1e, N_EROWS, HID, EIN_W, 0,
                                        nullptr, nullptr, edge_rep, cn);

  // 3. BN stats for layer 1
  stats_partial <<<STATS_G, HID, 0, stream>>>(Y1e, N_EROWS, HID, part);
  stats_finalize<<<1, HID, 0, stream>>>(part, HID, N_EROWS, ge1, be1, sc_e1, sh_e1);
  stats_partial <<<STATS_G, HID, 0, stream>>>(Y1c, N_CYC, HID, part);
  stats_finalize<<<1, HID, 0, stream>>>(part, HID, N_CYC, gc1, bc1, sc_c1, sh_c1);

  // 4. layer-2 GEMMs (BN+relu fused into A load), write pre-BN into d_out
  dim3 g2e((N_EROWS + 127) / 128, HCH / 64);
  gemm_wmma<2><<<g2e, blk, 0, stream>>>(Y1e, We2, outE, N_EROWS, HCH, HID, HID,
                                        sc_e1, sh_e1, nullptr, nullptr);
  dim3 g2c((N_CYC + 127) / 128, HCH / 64);
  gemm_wmma<2><<<g2c, blk, 0, stream>>>(Y1c, Wc2, outC, N_CYC, HCH, HID, HID,
                                        sc_c1, sh_c1, nullptr, nullptr);

  // 5. output BN stats + in-place bn_relu
  stats_partial <<<STATS_G, HCH, 0, stream>>>(outE, N_EROWS, HCH, part);
  stats_finalize<<<1, HCH, 0, stream>>>(part, HCH, N_EROWS, ge2, be2, sc_e2, sh_e2);
  stats_partial <<<STATS_G, HCH, 0, stream>>>(outC, N_CYC, HCH, part);
  stats_finalize<<<1, HCH, 0, stream>>>(part, HCH, N_CYC, gc2, bc2, sc_c2, sh_c2);

  size_t totE4 = (size_t)N_EROWS * HCH / 4, totC4 = (size_t)N_CYC * HCH / 4;
  bn_relu_apply<<<(unsigned)((totE4 + 255) / 256), 256, 0, stream>>>(outE, totE4, HCH, sc_e2, sh_e2);
  bn_relu_apply<<<(unsigned)((totC4 + 255) / 256), 256, 0, stream>>>(outC, totC4, HCH, sc_c2, sh_c2);
}